// GraphNeuralNetwork_43499428774207
// MI455X (gfx1250) — compile-verified
//
#include <hip/hip_runtime.h>
#include <math.h>

typedef float v2f __attribute__((ext_vector_type(2)));
typedef float v8f __attribute__((ext_vector_type(8)));

#define D_IN  256
#define D_OUT 128
#define N_GRAPHS 64
#define BROWS 32          // M rows per block (100000 % 32 == 0)
#define XPITCH 260        // LDS pitch: 260 mod 64 = 4 -> bank-conflict-free

// ---------------------------------------------------------------------------
// init: deg = 1.0 (self-loop weight), pooled/counts = 0
// ---------------------------------------------------------------------------
__global__ void k_init(float* __restrict__ deg, float* __restrict__ pooled,
                       float* __restrict__ counts, int n) {
    int i = blockIdx.x * blockDim.x + threadIdx.x;
    if (i < n) deg[i] = 1.0f;
    if (i < N_GRAPHS * D_OUT) pooled[i] = 0.0f;
    if (i < N_GRAPHS) counts[i] = 0.0f;
}

// deg[dst] += w  (edge part of segment_sum)
__global__ void k_degree(const int* __restrict__ ei, const float* __restrict__ w,
                         float* __restrict__ deg, int E, int n) {
    int e = blockIdx.x * blockDim.x + threadIdx.x;
    if (e >= E) return;
    int d = ei[E + e];
    if ((unsigned)d < (unsigned)n) atomicAdd(&deg[d], w[e]);
}

// dis = deg > 0 ? rsqrt(deg) : 0   (in place)
__global__ void k_rsqrt(float* __restrict__ deg, int n) {
    int i = blockIdx.x * blockDim.x + threadIdx.x;
    if (i >= n) return;
    float d = deg[i];
    deg[i] = (d > 0.0f) ? rsqrtf(d) : 0.0f;
}

// ---------------------------------------------------------------------------
// h = x @ W   via V_WMMA_F32_16X16X4_F32 (fp32 WMMA, fp32 accumulate)
// block = 256 threads = 8 waves; block tile = 32(M) x 128(N)
// each wave owns 16 N-cols and TWO 16x16 M-subtiles -> each B fragment load
// feeds 2 WMMAs (amortized); x tile (32x256 f32) staged in LDS.
// ---------------------------------------------------------------------------
__global__ void __launch_bounds__(256)
k_gemm_wmma(const float* __restrict__ x, const float* __restrict__ W,
            float* __restrict__ h, int n) {
    __shared__ float xs[BROWS * XPITCH];
    const int row0 = blockIdx.x * BROWS;

    // cooperative load of the 32x256 x-tile (float4 per transaction)
    for (int t = threadIdx.x; t < BROWS * (D_IN / 4); t += 256) {
        int r  = t >> 6;          // row 0..31
        int c4 = t & 63;          // float4 chunk 0..63
        float4 v = *(const float4*)(x + (size_t)(row0 + r) * D_IN + c4 * 4);
        *(float4*)(xs + r * XPITCH + c4 * 4) = v;
    }
    __syncthreads();

    const int lane  = threadIdx.x & 31;
    const int wave  = threadIdx.x >> 5;
    const int col0  = wave * 16;       // 8 waves cover all 128 output cols
    const int laneN = lane & 15;
    const int hi    = lane >> 4;       // lanes 16..31 carry K+2 / K+3

    const float* xrow0 = xs + (size_t)laneN * XPITCH;            // rows 0..15
    const float* xrow1 = xs + (size_t)(laneN + 16) * XPITCH;     // rows 16..31
    v8f acc0 = {};
    v8f acc1 = {};

#pragma unroll 4
    for (int k = 0; k < D_IN; k += 4) {
        __builtin_prefetch((const void*)(W + (size_t)(k + 16) * D_OUT + col0), 0, 1);
        v2f a0, a1, b;
        // B 4x16 f32: lanes 0-15 -> rows K=k,k+1 ; lanes 16-31 -> K=k+2,k+3
        b.x = W[(size_t)(k + 2 * hi + 0) * D_OUT + col0 + laneN];
        b.y = W[(size_t)(k + 2 * hi + 1) * D_OUT + col0 + laneN];
        // A 16x4 f32 fragments for both M-subtiles
        a0.x = xrow0[k + 2 * hi + 0];
        a0.y = xrow0[k + 2 * hi + 1];
        a1.x = xrow1[k + 2 * hi + 0];
        a1.y = xrow1[k + 2 * hi + 1];
        acc0 = __builtin_amdgcn_wmma_f32_16x16x4_f32(
            false, a0, false, b, (short)0, acc0, false, false);
        acc1 = __builtin_amdgcn_wmma_f32_16x16x4_f32(
            false, a1, false, b, (short)0, acc1, false, false);
    }

    // C/D layout: VGPR r -> M = r (+8 for lanes 16-31), N = lane&15
#pragma unroll
    for (int r = 0; r < 8; ++r) {
        int row = row0 + r + 8 * hi;
        h[(size_t)row * D_OUT + col0 + laneN] = acc0[r];
        h[(size_t)(row + 16) * D_OUT + col0 + laneN] = acc1[r];
    }
}

// out = h * dis^2   (self-loop term: norm = dis[i]*1*dis[i]; avoids atomics)
__global__ void k_selfloop(const float* __restrict__ h, const float* __restrict__ dis,
                           float* __restrict__ out, int n) {
    size_t idx = (size_t)blockIdx.x * blockDim.x + threadIdx.x;
    if (idx >= (size_t)n * D_OUT) return;
    int node = (int)(idx >> 7);
    float di = dis[node];
    out[idx] = h[idx] * di * di;
}

// out[dst] += norm * h[src] : one wave32 per edge, float4 gather, 4 atomics/lane
// h (51.2 MB) and out (51.2 MB) both fit in the 192 MB L2 -> gathers and
// atomics run out of L2, not HBM.
__global__ void k_edge_scatter(const int* __restrict__ ei, const float* __restrict__ w,
                               const float* __restrict__ dis, const float* __restrict__ h,
                               float* __restrict__ out, int E, int n) {
    size_t tid = (size_t)blockIdx.x * blockDim.x + threadIdx.x;
    int e    = (int)(tid >> 5);
    int lane = (int)(tid & 31);
    if (e >= E) return;
    int s = ei[e];
    int d = ei[E + e];
    if ((unsigned)s >= (unsigned)n || (unsigned)d >= (unsigned)n) return;
    float norm = dis[s] * w[e] * dis[d];
    float4 hv = *(const float4*)(h + (size_t)s * D_OUT + lane * 4);
    float* o = out + (size_t)d * D_OUT + lane * 4;
    atomicAdd(o + 0, hv.x * norm);
    atomicAdd(o + 1, hv.y * norm);
    atomicAdd(o + 2, hv.z * norm);
    atomicAdd(o + 3, hv.w * norm);
}

// pooled[batch[node]] += relu(out + b)
__global__ void k_relu_pool(const float* __restrict__ out, const float* __restrict__ b,
                            const int* __restrict__ batch, float* __restrict__ pooled,
                            int n) {
    size_t idx = (size_t)blockIdx.x * blockDim.x + threadIdx.x;
    if (idx >= (size_t)n * D_OUT) return;
    int node = (int)(idx >> 7);
    int j    = (int)(idx & (D_OUT - 1));
    float v = fmaxf(out[idx] + b[j], 0.0f);
    int g = batch[node] & (N_GRAPHS - 1);
    atomicAdd(&pooled[(size_t)g * D_OUT + j], v);
}

__global__ void k_counts(const int* __restrict__ batch, float* __restrict__ counts,
                         int n) {
    int i = blockIdx.x * blockDim.x + threadIdx.x;
    if (i >= n) return;
    atomicAdd(&counts[batch[i] & (N_GRAPHS - 1)], 1.0f);
}

// logits[g] = sigmoid( dot(pooled[g], Wlin)/count[g] + blin ) : one wave / graph
__global__ void k_final(const float* __restrict__ pooled, const float* __restrict__ counts,
                        const float* __restrict__ Wlin, const float* __restrict__ blin,
                        float* __restrict__ outp) {
    int g = blockIdx.x;
    int lane = threadIdx.x;
    float p = 0.0f;
    for (int j = lane; j < D_OUT; j += 32)
        p += pooled[(size_t)g * D_OUT + j] * Wlin[j];
    for (int off = 16; off > 0; off >>= 1)
        p += __shfl_down(p, off, 32);
    if (lane == 0) {
        float logit = p / fmaxf(counts[g], 1.0f) + blin[0];
        outp[g] = 1.0f / (1.0f + expf(-logit));
    }
}

// ---------------------------------------------------------------------------
extern "C" void kernel_launch(void* const* d_in, const int* in_sizes, int n_in,
                              void* d_out, int out_size, void* d_ws, size_t ws_size,
                              hipStream_t stream) {
    const float* x     = (const float*)d_in[0];
    const int*   ei    = (const int*)d_in[1];   // [2, E] row-major: src then dst
    const float* ew    = (const float*)d_in[2];
    const int*   batch = (const int*)d_in[3];
    const float* Wc    = (const float*)d_in[4]; // [256,128] row-major
    const float* bc    = (const float*)d_in[5];
    const float* Wl    = (const float*)d_in[6]; // [128,1]
    const float* bl    = (const float*)d_in[7];
    float* outp = (float*)d_out;

    const int N = in_sizes[3];   // 100000 (batch length)
    const int E = in_sizes[2];   // 1600000 (edge_weight length)

    // workspace layout (floats): deg/dis[N] | pooled[64*128] | counts[64] | h[N*128] | out[N*128]
    float* ws     = (float*)d_ws;
    float* deg    = ws;                                   // becomes dis after k_rsqrt
    float* pooled = ws + N;
    float* counts = pooled + (size_t)N_GRAPHS * D_OUT;
    size_t off_h  = ((size_t)N + N_GRAPHS * D_OUT + N_GRAPHS + 127) & ~(size_t)127;
    float* h      = ws + off_h;
    float* outacc = h + (size_t)N * D_OUT;
    // total: ~103 MB of scratch

    const size_t nf = (size_t)N * D_OUT;

    k_init<<<(N + 255) / 256, 256, 0, stream>>>(deg, pooled, counts, N);
    k_degree<<<(E + 255) / 256, 256, 0, stream>>>(ei, ew, deg, E, N);
    k_rsqrt<<<(N + 255) / 256, 256, 0, stream>>>(deg, N);
    k_gemm_wmma<<<N / BROWS, 256, 0, stream>>>(x, Wc, h, N);       // N % 32 == 0
    k_selfloop<<<(int)((nf + 255) / 256), 256, 0, stream>>>(h, deg, outacc, N);
    k_edge_scatter<<<(int)(((size_t)E * 32 + 255) / 256), 256, 0, stream>>>(
        ei, ew, deg, h, outacc, E, N);
    k_relu_pool<<<(int)((nf + 255) / 256), 256, 0, stream>>>(outacc, bc, batch, pooled, N);
    k_counts<<<(N + 255) / 256, 256, 0, stream>>>(batch, counts, N);
    k_final<<<N_GRAPHS, 32, 0, stream>>>(pooled, counts, Wl, bl, outp);
}